// CardEmbedding_14096082666288
// MI455X (gfx1250) — compile-verified
//
#include <hip/hip_runtime.h>
#include <cstdint>

// ---- problem constants (from reference) ----
#define RMIN    256
#define RMAX    1280
#define IN_DIM  2048
#define EMB     18
#define BATCH   8192

// ---- tiling ----
#define BLOCK      256                      // 8 wave32 per block
#define TILE_RECS  512                      // records per block (2 per thread)
#define TILE_F4    (TILE_RECS * EMB / 4)    // 2304 float4 = 36,864 B staged in LDS

// native clang vector types (work with nontemporal/amdgcn builtins)
typedef float f32x2 __attribute__((ext_vector_type(2)));
typedef float f32x4 __attribute__((ext_vector_type(4)));
typedef int   v4i   __attribute__((ext_vector_type(4)));
typedef __attribute__((address_space(3))) v4i lds_v4i;

#if defined(__gfx1250__) &&                                                   \
    __has_builtin(__builtin_amdgcn_global_store_async_from_lds_b128) &&       \
    __has_builtin(__builtin_amdgcn_s_wait_asynccnt)
#define USE_ASYNC_STORE 1
#else
#define USE_ASYNC_STORE 0
#endif

// gfx12-style cpol immediate: TH in low bits; store TH_NT == 1 (ISA §7.3).
// Output is a 1.2 GB write-once stream (6x the 192 MB L2) -> non-temporal.
#define CPOL_STORE_NT 1

__global__ __launch_bounds__(BLOCK) void card_embed_kernel(
    const float* __restrict__ x, const float* __restrict__ table,
    float* __restrict__ out)
{
    __shared__ f32x4 s_tile[TILE_F4];       // 36,864 bytes

    const int tid  = threadIdx.x;
    const int blk  = blockIdx.x;            // 4 tiles per batch row (512 | 2048)
    const int b    = blk >> 2;
    const int posb = (blk & 3) * TILE_RECS;
    const int pos0 = posb + tid * 2;        // this thread's 2 consecutive records

    // warm L2 with the (tiny, permanently hot) table
    __builtin_prefetch(table, 0, 3);

    // stream-load both x values for this pair (8B-aligned: pos0 is even)
    const f32x2 xv = __builtin_nontemporal_load(
        (const f32x2*)(x + (size_t)b * IN_DIM + pos0));

    // build 2 records = 36 floats = 9 float4 in registers
    f32x4  r[9];
    float* rf  = (float*)r;
    f32x2* rf2 = (f32x2*)r;

    // region boundaries (256/1280) are multiples of 64 => wave-uniform branch
    if (pos0 >= RMIN && pos0 < RMAX) {
        const int i0 = (int)xv.x;
        const int i1 = (int)xv.y;
        const f32x2* t0 = (const f32x2*)(table + i0 * EMB);  // 72B rows, 8B-aligned
        const f32x2* t1 = (const f32x2*)(table + i1 * EMB);
        #pragma unroll
        for (int k = 0; k < 9; ++k) rf2[k] = t0[k];
        #pragma unroll
        for (int k = 0; k < 9; ++k) rf2[9 + k] = t1[k];
    } else {
        #pragma unroll
        for (int k = 0; k < 18; ++k) rf[k] = xv.x;
        #pragma unroll
        for (int k = 0; k < 18; ++k) rf[18 + k] = xv.y;
    }

    // stage to LDS: pair starts at 144B boundary -> 9 aligned b128 DS stores
    #pragma unroll
    for (int k = 0; k < 9; ++k) s_tile[tid * 9 + k] = r[k];

    __syncthreads();   // ds writes drained (DScnt) before async engine reads LDS

    // stream tile out: fully coalesced, 16B-aligned (tile base = blk * 36,864 B)
    f32x4* outp = (f32x4*)out + (size_t)blk * TILE_F4;

#if USE_ASYNC_STORE
    // ASYNCcnt-tracked LDS->global DMA stores: one wave32 b128 op moves 512
    // contiguous bytes, no VALU/VGPR involvement in the data path.
    #pragma unroll
    for (int k = 0; k < 9; ++k) {
        const int j = tid + k * BLOCK;
        __builtin_amdgcn_global_store_async_from_lds_b128(
            (v4i*)(outp + j),
            (lds_v4i*)(uintptr_t)(s_tile + j),
            0, CPOL_STORE_NT);
    }
    __builtin_amdgcn_s_wait_asynccnt(0);
#else
    #pragma unroll
    for (int k = 0; k < 9; ++k) {
        const int j = tid + k * BLOCK;
        __builtin_nontemporal_store(s_tile[j], &outp[j]);   // TH=NT write-once stream
    }
#endif
}

extern "C" void kernel_launch(void* const* d_in, const int* in_sizes, int n_in,
                              void* d_out, int out_size, void* d_ws, size_t ws_size,
                              hipStream_t stream) {
    const float* x     = (const float*)d_in[0];   // (8192, 1, 2048) f32 (int values)
    const float* table = (const float*)d_in[1];   // (512, 18) f32
    float*       out   = (float*)d_out;           // (8192, 2048, 18) f32

    const int n_blocks = (BATCH * IN_DIM) / TILE_RECS;   // 32,768
    card_embed_kernel<<<dim3(n_blocks), dim3(BLOCK), 0, stream>>>(x, table, out);
}